// COCOMatchingAlgorithm_20444044329049
// MI455X (gfx1250) — compile-verified
//
#include <hip/hip_runtime.h>
#include <stdint.h>

#define BD 2048    // detections per batch
#define GG 256     // ground truths per batch
#define TT 10      // IoU thresholds
#define NTHREADS 1024

__device__ const float c_thr[TT] = {0.5f, 0.55f, 0.6f, 0.65f, 0.7f,
                                    0.75f, 0.8f, 0.85f, 0.9f, 0.95f};

__global__ void zero_kernel(float* __restrict__ out, int n) {
  int i = blockIdx.x * blockDim.x + threadIdx.x;
  if (i < n) out[i] = 0.0f;
}

// Async-copy one 256-float IoU row (1KB) into LDS: two
// GLOBAL_LOAD_ASYNC_TO_LDS_B128 per wave (32 lanes x 16B = 512B each).
// Per CDNA5 ISA the instruction offset is added to BOTH the LDS
// and the global address, so one address pair + offset:512 covers the row.
__device__ __forceinline__ void async_row_b128(uint32_t lds_base, const float* row, int lane) {
  uint64_t ga = (uint64_t)(uintptr_t)row + (uint64_t)(lane * 16);
  uint32_t la = lds_base + (uint32_t)(lane * 16);
  asm volatile("global_load_async_to_lds_b128 %0, %1, off" :: "v"(la), "v"(ga) : "memory");
  asm volatile("global_load_async_to_lds_b128 %0, %1, off offset:512" :: "v"(la), "v"(ga) : "memory");
}

__global__ __launch_bounds__(NTHREADS) void coco_match_kernel(
    const float* __restrict__ ious,       // (B, D, G)
    const int*   __restrict__ det_cls,    // (B, D)
    const float* __restrict__ det_score,  // (B, D)
    const int*   __restrict__ gt_cls,     // (B, G)
    float* __restrict__ out_det,          // (B, D, T)
    float* __restrict__ out_gt)           // (B, G, T)
{
  __shared__ float s_score[BD];                        // 8 KB
  __shared__ int   s_sidx[BD];                         // 8 KB
  __shared__ int   s_cls[BD];                          // 8 KB
  __shared__ __align__(16) float s_stage[TT][2][GG];   // 20 KB double-buffered rows

  const int b   = blockIdx.x;
  const int tid = threadIdx.x;

  // ---------- Phase 1: stable bitonic sort by score desc (index asc on ties) ----------
  for (int i = tid; i < BD; i += NTHREADS) {
    s_score[i] = det_score[(size_t)b * BD + i];
    s_sidx[i]  = i;
  }
  __syncthreads();

  for (int k = 2; k <= BD; k <<= 1) {
    for (int j = k >> 1; j > 0; j >>= 1) {
      for (int i = tid; i < BD; i += NTHREADS) {
        int ixj = i ^ j;
        if (ixj > i) {
          float sa = s_score[i], sb = s_score[ixj];
          int   ia = s_sidx[i],  ib = s_sidx[ixj];
          bool b_before_a = (sb > sa) || ((sb == sa) && (ib < ia));
          bool a_before_b = (sa > sb) || ((sa == sb) && (ia < ib));
          bool desc = ((i & k) == 0);
          if (desc ? b_before_a : a_before_b) {
            s_score[i] = sb; s_score[ixj] = sa;
            s_sidx[i]  = ib; s_sidx[ixj]  = ia;
          }
        }
      }
      __syncthreads();
    }
  }

  for (int i = tid; i < BD; i += NTHREADS) {
    s_cls[i] = det_cls[(size_t)b * BD + s_sidx[i]];
  }
  __syncthreads();

  // ---------- Phase 2: greedy matching, one wave per IoU threshold ----------
  const int wave = tid >> 5;
  const int lane = tid & 31;
  if (wave >= TT) return;   // surplus sort waves terminate (no barriers remain)

  const float thr = c_thr[wave];

  // Lane owns 8 contiguous GTs [8*lane, 8*lane+8) -> aligned b128 LDS reads.
  int gc[8];
  #pragma unroll
  for (int k = 0; k < 8; ++k)
    gc[k] = gt_cls[(size_t)b * GG + (lane << 3) + k];

  uint32_t matched = 0u;  // bit k <=> GT (8*lane + k) already matched at this threshold

  const uint32_t stage0 = (uint32_t)(uintptr_t)(&s_stage[wave][0][0]);
  const uint32_t stage1 = (uint32_t)(uintptr_t)(&s_stage[wave][1][0]);

  // Prime the pipeline.
  {
    const float* row0 = ious + ((size_t)b * BD + s_sidx[0]) * GG;
    async_row_b128(stage0, row0, lane);
  }

  for (int d = 0; d < BD; ++d) {
    const int cb = d & 1;
    if (d + 1 < BD) {
      const float* rown = ious + ((size_t)b * BD + s_sidx[d + 1]) * GG;
      async_row_b128(cb ? stage0 : stage1, rown, lane);
      asm volatile("s_wait_asynccnt 0x2" ::: "memory");  // row d has landed
    } else {
      asm volatile("s_wait_asynccnt 0x0" ::: "memory");
    }

    const int c = s_cls[d];
    const float4* st4 = reinterpret_cast<const float4*>(&s_stage[wave][cb][lane << 3]);
    float4 q0 = st4[0];
    float4 q1 = st4[1];
    float vv[8] = {q0.x, q0.y, q0.z, q0.w, q1.x, q1.y, q1.z, q1.w};

    // Monotonic packed key: (iou_bits << 32) | (255 - g).  IoUs are positive so
    // float bits are order-preserving; low field => ties pick smallest g.
    uint64_t key = 0ull;
    #pragma unroll
    for (int k = 0; k < 8; ++k) {
      int g = (lane << 3) + k;
      bool valid = (gc[k] == c) && (gc[k] > 0) && (vv[k] > thr) &&
                   (((matched >> k) & 1u) == 0u);
      uint64_t kk = ((uint64_t)__float_as_uint(vv[k]) << 32) | (uint32_t)(255 - g);
      if (valid && kk > key) key = kk;
    }

    // Wave-uniform early out: nothing matched this detection.
    if (__ballot(key != 0ull) == 0ull) continue;

    // 5-step wave32 max-reduction on the packed key.
    #pragma unroll
    for (int off = 16; off > 0; off >>= 1) {
      uint32_t lo = (uint32_t)key, hi = (uint32_t)(key >> 32);
      uint32_t olo = __shfl_xor(lo, off, 32);
      uint32_t ohi = __shfl_xor(hi, off, 32);
      uint64_t ok = ((uint64_t)ohi << 32) | olo;
      if (ok > key) key = ok;
    }

    const int bg = 255 - (int)(key & 0xFFFFFFFFull);
    if ((bg >> 3) == lane) {
      matched |= 1u << (bg & 7);
      out_gt[((size_t)b * GG + bg) * TT + wave] = 1.0f;
    }
    if (lane == 0) {
      out_det[((size_t)b * BD + s_sidx[d]) * TT + wave] = 1.0f;
    }
  }
}

extern "C" void kernel_launch(void* const* d_in, const int* in_sizes, int n_in,
                              void* d_out, int out_size, void* d_ws, size_t ws_size,
                              hipStream_t stream) {
  (void)n_in; (void)d_ws; (void)ws_size;
  const float* iousp  = (const float*)d_in[0];  // (B, D, G) f32
  const int*   dcls   = (const int*)  d_in[1];  // (B, D)    i32
  const float* dscore = (const float*)d_in[2];  // (B, D)    f32
  const int*   gcls   = (const int*)  d_in[3];  // (B, G)    i32
  float* out = (float*)d_out;

  const int B = in_sizes[1] / BD;
  float* out_det = out;                          // (B, D, T)
  float* out_gt  = out + (size_t)B * BD * TT;    // (B, G, T)

  zero_kernel<<<(out_size + 255) / 256, 256, 0, stream>>>(out, out_size);
  coco_match_kernel<<<B, NTHREADS, 0, stream>>>(iousp, dcls, dscore, gcls, out_det, out_gt);
}